// MultiresEnc_82403242541324
// MI455X (gfx1250) — compile-verified
//
#include <hip/hip_runtime.h>
#include <math.h>
#include <stdint.h>

// Multires hash-grid encoding (2D, 16 levels, 2^16-slot tables, 2 features).
// Gather-bound kernel: tables (8 MB total) are L2-resident; HBM traffic is
// ~136 MB (x in + out) -> ~6 us floor at 23.3 TB/s.
// CDNA5 paths used: GLOBAL_STORE_ASYNC_FROM_LDS_B128 (non-temporal) for
// coalesced output (LDS -> global DMA, no VGPR round trip) + s_wait_asynccnt.
// v2 -> v3: weights computed as frac(x*N) / 1-frac(x*N)  (algebraically equal
// to the reference's (vmax-x)/denom within ~1 ulp), and the bilinear blend
// expressed as float2 vectors so it lowers to packed v_pk_mul/v_pk_fma_f32.
// Cuts per-level VALU from ~40 to ~20 wave-ops so instruction issue no longer
// exceeds the HBM roofline.

#define NUM_LEVELS 16
#define TABLE      65536
#define PRIME1     19349663u
#define BLOCK      256
#define ROWF       (NUM_LEVELS * 2)   // 32 floats (=128 B, one cacheline) per point
#define LDS_STRIDE 34                 // padded row stride (floats): 8B-aligned rows,
                                      // conflict-free ds stores across a wave

typedef float v2f __attribute__((ext_vector_type(2)));

struct LevelConsts {
    float Nf[NUM_LEVELS];    // per-level resolution, f32 (matches reference floor(16*b^l))
};

__global__ __launch_bounds__(BLOCK) void
MultiresEnc_82403242541324_kernel(const float* __restrict__ x,
                                  const float* __restrict__ table,
                                  float* __restrict__ out,
                                  LevelConsts lc)
{
    __shared__ float sm[BLOCK * LDS_STRIDE];

    const int t   = threadIdx.x;
    const int gid = blockIdx.x * BLOCK + t;

    // Coalesced b64 load of this thread's point.
    const float2 p  = ((const float2*)x)[gid];
    const float  px = p.x, py = p.y;

    float* row = &sm[t * LDS_STRIDE];

#pragma unroll
    for (int l = 0; l < NUM_LEVELS; ++l) {
        const float Nf = lc.Nf[l];

        // Scaled coords; weights via fractional part (packs per axis):
        //   w2 = frac(x*N) == (x - vmin)/denom, w1 = 1 - w2   (to ~1 ulp)
        const float tx  = px * Nf;            // pk_mul across x/y
        const float ty  = py * Nf;
        const float flx = floorf(tx);
        const float fly = floorf(ty);
        const float w2x = tx - flx, w2y = ty - fly;
        const float w1x = 1.0f - w2x, w1y = 1.0f - w2y;

        // Spatial hash: (x*1) ^ (y*PRIME1) mod 2^16
        // (int32 wraparound == uint mul; &0xFFFF == mod; xor+and fuse to v_bitop3)
        const int x0 = (int)flx, y0 = (int)fly;
        const unsigned hx0 = (unsigned)x0;
        const unsigned hx1 = (unsigned)(x0 + 1);
        const unsigned hy0 = (unsigned)y0 * PRIME1;
        const unsigned hy1 = hy0 + PRIME1;            // (y0+1)*PRIME1 mod 2^32
        const unsigned h00 = (hx0 ^ hy0) & 0xFFFFu;   // vertex (0,0)
        const unsigned h01 = (hx0 ^ hy1) & 0xFFFFu;   // vertex (0,1)
        const unsigned h10 = (hx1 ^ hy0) & 0xFFFFu;   // vertex (1,0)
        const unsigned h11 = (hx1 ^ hy1) & 0xFFFFu;   // vertex (1,1)

        // 4 gathers of float2 (global_load_b64); tables are L2-resident.
        const v2f* tab = (const v2f*)table + (size_t)l * TABLE;
        const v2f f00 = tab[h00];
        const v2f f01 = tab[h01];
        const v2f f10 = tab[h10];
        const v2f f11 = tab[h11];

        // Bilinear blend on packed f32 pairs (v_pk_mul_f32 / v_pk_fma_f32):
        // f1 = w1x*v0 + w2x*v2 ; f2 = w1x*v1 + w2x*v3 ; out = w1y*f1 + w2y*f2
        const v2f f1 = w1x * f00 + w2x * f10;
        const v2f f2 = w1x * f01 + w2x * f11;
        const v2f o  = w1y * f1  + w2y * f2;

        *(v2f*)&row[2 * l] = o;               // 8B-aligned ds store
    }

    __syncthreads();

    // Coalesced output: each wave instruction stores 32 lanes x 16 B = 512 B
    // of contiguous global memory straight out of LDS (async DMA engine).
    // Non-temporal: the 128 MB output stream is never re-read; keep it from
    // competing with the 8 MB of hot hash tables in the near caches.
    float* outBlock = out + (size_t)blockIdx.x * (BLOCK * ROWF);
#pragma unroll
    for (int i = 0; i < 8; ++i) {
        const unsigned gf   = (unsigned)i * (BLOCK * 4) + (unsigned)t * 4; // float idx in 8192-float tile
        const unsigned pt   = gf >> 5;                                     // point within block
        const unsigned j    = gf & 31u;                                    // element within row
        const unsigned loff = (unsigned)(uintptr_t)(&sm[pt * LDS_STRIDE + j]); // LDS byte addr (low 32b of flat)
        const unsigned goff = gf * 4u;                                     // byte offset within tile
        asm volatile("global_store_async_from_lds_b128 %0, %1, %2 th:TH_STORE_NT"
                     :: "v"(goff), "v"(loff), "s"(outBlock)
                     : "memory");
    }
    // Drain the async engine before wave teardown (S_ENDPGM also wait-idles,
    // but make the dependency explicit).
    asm volatile("s_wait_asynccnt 0" ::: "memory");
}

extern "C" void kernel_launch(void* const* d_in, const int* in_sizes, int n_in,
                              void* d_out, int out_size, void* d_ws, size_t ws_size,
                              hipStream_t stream) {
    (void)n_in; (void)d_ws; (void)ws_size; (void)out_size;

    const float* x    = (const float*)d_in[0];   // [P,2] f32
    const float* kern = (const float*)d_in[1];   // [16,65536,2] f32
    float*       out  = (float*)d_out;           // [P,32] f32

    // Per-level resolutions in f32, mimicking the reference:
    // b = exp((ln 1024 - ln 16)/15); N_l = floor(16 * b^l), l = 1..16
    LevelConsts lc;
    const float b = expf((logf(1024.0f) - logf(16.0f)) / 15.0f);
    for (int l = 0; l < NUM_LEVELS; ++l) {
        lc.Nf[l] = floorf(16.0f * powf(b, (float)(l + 1)));
    }

    const int P       = in_sizes[0] / 2;   // 1048576
    const int nblocks = P / BLOCK;         // divisible: 4096 blocks

    MultiresEnc_82403242541324_kernel<<<nblocks, BLOCK, 0, stream>>>(x, kern, out, lc);
}